// Attn_49632642073030
// MI455X (gfx1250) — compile-verified
//
#include <hip/hip_runtime.h>
#include <hip/hip_bf16.h>

#define H 4096
#define S 8192

typedef float v2f __attribute__((ext_vector_type(2)));
typedef float v8f __attribute__((ext_vector_type(8)));
typedef unsigned int v4u __attribute__((ext_vector_type(4)));
typedef int v4i __attribute__((ext_vector_type(4)));
typedef int v8i __attribute__((ext_vector_type(8)));

#define TILE_ROWS 128           // rows per workgroup (8 waves x 16)
#define TILE_COLS 64            // K columns per TDM tile
#define LDS_STRIDE 68           // 64 data + 4 pad DWORDs (bank-conflict-free, 8B aligned)
#define NT 16                   // tiles per 1024-column K chunk

// ---------------------------------------------------------------------------
// Phase 1a: partial v = W^T h over 256-row slabs (coalesced 1KB reads of W).
// ---------------------------------------------------------------------------
__global__ __launch_bounds__(256) void wTh_partial(const float* __restrict__ W,
                                                   const float* __restrict__ h,
                                                   float* __restrict__ vpart) {
    const int c  = blockIdx.x * 256 + threadIdx.x;
    const int r0 = blockIdx.y * 256;
    float acc = 0.0f;
#pragma unroll 8
    for (int r = 0; r < 256; ++r) {
        acc = fmaf(W[(size_t)(r0 + r) * H + c], h[r0 + r], acc);
    }
    vpart[(size_t)blockIdx.y * H + c] = acc;
}

// ---------------------------------------------------------------------------
// Phase 1b: v[c] = sum of 16 partials.
// ---------------------------------------------------------------------------
__global__ __launch_bounds__(256) void reduce_v(const float* __restrict__ vpart,
                                                float* __restrict__ v) {
    const int c = blockIdx.x * 256 + threadIdx.x;
    float s = 0.0f;
#pragma unroll
    for (int i = 0; i < 16; ++i) s += vpart[(size_t)i * H + c];
    v[c] = s;
}

// ---------------------------------------------------------------------------
// TDM: issue one TENSOR_LOAD_TO_LDS of a TILE_ROWS x TILE_COLS f32 tile.
// D# packing per CDNA5 ISA ch.8 (2D, data_size=4B, LDS row padding 4 DWORDs
// every 64 DWORDs -> LDS stride 68 floats).
// ---------------------------------------------------------------------------
__device__ __forceinline__ void tdm_load_tile(const float* gptr, unsigned lds_off) {
    const unsigned long long ga = (unsigned long long)(uintptr_t)gptr;
    v4u g0;
    g0[0] = 1u;                                            // count=1, user mode
    g0[1] = lds_off;                                       // lds_addr
    g0[2] = (unsigned)ga;                                  // global_addr[31:0]
    g0[3] = ((unsigned)(ga >> 32) & 0x01FFFFFFu)           // global_addr[56:32]
          | (2u << 30);                                    // type=2 ("image")
    v8i g1;
    g1[0] = (int)((2u << 16)      // data_size = 4 bytes
                | (1u << 20)      // pad_enable
                | (5u << 22)      // pad_interval: every 64 DWORDs
                | (3u << 25));    // pad_amount: 4 DWORDs
    g1[1] = (int)(((unsigned)H & 0xFFFFu) << 16);          // tensor_dim0 lo
    g1[2] = (int)((((unsigned)H >> 16) & 0xFFFFu)          // tensor_dim0 hi
                | (((unsigned)S & 0xFFFFu) << 16));        // tensor_dim1 lo
    g1[3] = (int)((((unsigned)S >> 16) & 0xFFFFu)          // tensor_dim1 hi
                | ((unsigned)TILE_COLS << 16));            // tile_dim0
    g1[4] = TILE_ROWS;                                     // tile_dim1 (tile_dim2=0)
    g1[5] = H;                                             // tensor_dim0_stride lo
    g1[6] = 0;
    g1[7] = 0;
    v4i z4 = (v4i)0;
#if defined(__clang_major__) && (__clang_major__ >= 23)
    v8i z8 = (v8i)0;
    __builtin_amdgcn_tensor_load_to_lds(g0, g1, z4, z4, z8, 0);
#else
    __builtin_amdgcn_tensor_load_to_lds(g0, g1, z4, z4, 0);
#endif
}

// ---------------------------------------------------------------------------
// Phase 2: energies = enc @ v.  TDM double-buffers 128x64 f32 tiles into LDS;
// 8 waves feed V_WMMA_F32_16X16X4_F32 from LDS (conflict-free ds_load_b64).
//   grid (S/128 row-tiles, 4 K-splits), block 256 = 8 waves.
// ---------------------------------------------------------------------------
__global__ __launch_bounds__(256) void gemv_wmma(const float* __restrict__ enc,
                                                 const float* __restrict__ v,
                                                 float* __restrict__ epart) {
    __shared__ float tile[2][TILE_ROWS * LDS_STRIDE];

    const int lane = threadIdx.x & 31;
    const int wave = threadIdx.x >> 5;
    const int m    = lane & 15;
    const int hi   = lane >> 4;          // 0 or 1
    const int koff = hi * 2;             // K sub-offset for this lane half
    const int row0 = blockIdx.x * TILE_ROWS;
    const int k0   = blockIdx.y * (NT * TILE_COLS);   // 1024-column K chunk
    const bool issuer = (threadIdx.x < 32);

    // Prime the pipeline: tile 0 -> buffer 0.
    if (issuer) {
        tdm_load_tile(enc + (size_t)row0 * H + k0,
                      (unsigned)(uintptr_t)&tile[0][0]);
    }

    v8f acc = {};
    const int lds_row = (wave * 16 + m) * LDS_STRIDE;

    for (int t = 0; t < NT; ++t) {
        if (issuer) {
            if (t + 1 < NT) {   // prefetch next tile into the other buffer
                tdm_load_tile(enc + (size_t)row0 * H + k0 + (t + 1) * TILE_COLS,
                              (unsigned)(uintptr_t)&tile[(t + 1) & 1][0]);
                __builtin_amdgcn_s_wait_tensorcnt(1);   // tile t resident
            } else {
                __builtin_amdgcn_s_wait_tensorcnt(0);
            }
        }
        __syncthreads();

        const float* tb = &tile[t & 1][0];
        const int kt = k0 + t * TILE_COLS;
#pragma unroll 4
        for (int kk = 0; kk < TILE_COLS; kk += 4) {
            v2f a;
            {   // ds_load_b64: banks 4m / 4m+2 -> conflict-free
                const float2 ld = *(const float2*)(tb + lds_row + kk + koff);
                a[0] = ld.x; a[1] = ld.y;
            }
            // B column 0 = v[k..k+3]; branchless select keeps EXEC all-ones.
            const float2 vv = *(const float2*)(v + kt + kk + koff);
            v2f b;
            b[0] = (m == 0) ? vv.x : 0.0f;
            b[1] = (m == 0) ? vv.y : 0.0f;
            acc = __builtin_amdgcn_wmma_f32_16x16x4_f32(
                /*neg_a=*/false, a, /*neg_b=*/false, b,
                /*c_mod=*/(short)0, acc, /*reuse_a=*/false, /*reuse_b=*/false);
        }
        __syncthreads();
    }

    // Column N=0 of D: lane 0 -> M=0..7 in acc[0..7], lane 16 -> M=8..15.
    if (m == 0) {
        float* dst = epart + (size_t)blockIdx.y * S
                           + row0 + wave * 16 + hi * 8;
#pragma unroll
        for (int i = 0; i < 8; ++i) dst[i] = acc[i];
    }
}

// ---------------------------------------------------------------------------
// Phase 3: fold 4 K-split partials, softmax over S=8192 (one 1024-thr block).
// (b.h bias dropped: softmax is shift-invariant.)
// ---------------------------------------------------------------------------
__global__ __launch_bounds__(1024) void softmax_k(const float* __restrict__ epart,
                                                  float* __restrict__ out) {
    __shared__ float se[S];
    __shared__ float red[1024];
    const int t = threadIdx.x;

    float lmax = -INFINITY;
#pragma unroll
    for (int i = 0; i < S / 1024; ++i) {
        const int s = t + i * 1024;
        float e = epart[s] + epart[S + s] + epart[2 * S + s] + epart[3 * S + s];
        se[s] = e;
        lmax  = fmaxf(lmax, e);
    }
    red[t] = lmax;
    __syncthreads();
    for (int off = 512; off > 0; off >>= 1) {
        if (t < off) red[t] = fmaxf(red[t], red[t + off]);
        __syncthreads();
    }
    const float mx = red[0];
    __syncthreads();

    float lsum = 0.0f;
#pragma unroll
    for (int i = 0; i < S / 1024; ++i) lsum += expf(se[t + i * 1024] - mx);
    red[t] = lsum;
    __syncthreads();
    for (int off = 512; off > 0; off >>= 1) {
        if (t < off) red[t] += red[t + off];
        __syncthreads();
    }
    const float inv = 1.0f / red[0];

#pragma unroll
    for (int i = 0; i < S / 1024; ++i) {
        const int s = t + i * 1024;
        out[s] = expf(se[s] - mx) * inv;
    }
}

// ---------------------------------------------------------------------------
extern "C" void kernel_launch(void* const* d_in, const int* in_sizes, int n_in,
                              void* d_out, int out_size, void* d_ws, size_t ws_size,
                              hipStream_t stream) {
    const float* h   = (const float*)d_in[0];   // [1,1,H]
    const float* enc = (const float*)d_in[1];   // [1,S,H]
    const float* W   = (const float*)d_in[2];   // [H,H]
    // d_in[3] (bias b) is provably unused: softmax(e + b.h) == softmax(e).

    float* ws    = (float*)d_ws;
    float* vpart = ws;                          // 16*H   = 65536 f32
    float* v     = vpart + 16 * H;              // H      =  4096 f32
    float* epart = v + H;                       // 4*S    = 32768 f32  (~400KB total)

    wTh_partial<<<dim3(16, 16), 256, 0, stream>>>(W, h, vpart);
    reduce_v   <<<dim3(16),     256, 0, stream>>>(vpart, v);
    gemv_wmma  <<<dim3(S / TILE_ROWS, 4), 256, 0, stream>>>(enc, v, epart);
    softmax_k  <<<dim3(1), 1024, 0, stream>>>(epart, (float*)d_out);
}